// Memnet_88699664597679
// MI455X (gfx1250) — compile-verified
//
#include <hip/hip_runtime.h>
#include <hip/hip_bf16.h>
#include <math.h>

// ---- Problem constants (fixed by the reference) ----
#define NWORDS 32000
#define EDIM 512
#define WPI 4            // words per item
#define IPB 256          // items per batch
#define BATCHES 256
#define NITEMS (BATCHES * IPB)   // 65536
#define NHOPS 3
#define NACT 64

typedef _Float16 h8 __attribute__((ext_vector_type(8)));
typedef _Float16 h2 __attribute__((ext_vector_type(2)));
typedef float v2f __attribute__((ext_vector_type(2)));
typedef float v8f __attribute__((ext_vector_type(8)));

__device__ __forceinline__ float wave_sum(float v) {
    #pragma unroll
    for (int off = 16; off > 0; off >>= 1) v += __shfl_down(v, off, 32);
    return v;
}

// ---------------------------------------------------------------------------
// Kernel 1: EmbeddingBag(mean over 4 words) for A and C tables -> f16 tables.
// One wave per item; lane owns 16 consecutive dims (512 / 32).
// ---------------------------------------------------------------------------
__global__ void embed_bag_kernel(const int* __restrict__ x,
                                 const float* __restrict__ Aw,
                                 const float* __restrict__ Cw,
                                 _Float16* __restrict__ embA,
                                 _Float16* __restrict__ embC) {
    const int wave = threadIdx.x >> 5;
    const int lane = threadIdx.x & 31;
    const int item = blockIdx.x * 8 + wave;
    if (item >= NITEMS) return;

    int w[WPI];
    #pragma unroll
    for (int j = 0; j < WPI; ++j) w[j] = x[item * WPI + j];

    const int dbase = lane * 16;   // 16 dims per lane

    // Table A then table C (keeps register pressure low)
    for (int t = 0; t < 2; ++t) {
        const float* __restrict__ tab = (t == 0) ? Aw : Cw;
        _Float16* __restrict__ dst = (t == 0) ? embA : embC;
        float4 acc0 = {0,0,0,0}, acc1 = {0,0,0,0}, acc2 = {0,0,0,0}, acc3 = {0,0,0,0};
        #pragma unroll
        for (int j = 0; j < WPI; ++j) {
            const float4* p = reinterpret_cast<const float4*>(tab + (size_t)w[j] * EDIM + dbase);
            float4 a = p[0], b = p[1], c = p[2], d = p[3];
            acc0.x += a.x; acc0.y += a.y; acc0.z += a.z; acc0.w += a.w;
            acc1.x += b.x; acc1.y += b.y; acc1.z += b.z; acc1.w += b.w;
            acc2.x += c.x; acc2.y += c.y; acc2.z += c.z; acc2.w += c.w;
            acc3.x += d.x; acc3.y += d.y; acc3.z += d.z; acc3.w += d.w;
        }
        const float s = 0.25f;
        h8 lo, hi;
        lo[0]=(_Float16)(acc0.x*s); lo[1]=(_Float16)(acc0.y*s);
        lo[2]=(_Float16)(acc0.z*s); lo[3]=(_Float16)(acc0.w*s);
        lo[4]=(_Float16)(acc1.x*s); lo[5]=(_Float16)(acc1.y*s);
        lo[6]=(_Float16)(acc1.z*s); lo[7]=(_Float16)(acc1.w*s);
        hi[0]=(_Float16)(acc2.x*s); hi[1]=(_Float16)(acc2.y*s);
        hi[2]=(_Float16)(acc2.z*s); hi[3]=(_Float16)(acc2.w*s);
        hi[4]=(_Float16)(acc3.x*s); hi[5]=(_Float16)(acc3.y*s);
        hi[6]=(_Float16)(acc3.z*s); hi[7]=(_Float16)(acc3.w*s);
        h8* out = reinterpret_cast<h8*>(dst + (size_t)item * EDIM + dbase);
        out[0] = lo; out[1] = hi;
    }
}

// ---------------------------------------------------------------------------
// Kernel 2: u[b] = sum over 256 items of embC (segment_sum init).
// One block per batch; thread t owns dim t.
// ---------------------------------------------------------------------------
__global__ void u_init_kernel(const _Float16* __restrict__ embC,
                              float* __restrict__ u) {
    const int b = blockIdx.x;
    const int d = threadIdx.x;   // 0..511
    const _Float16* base = embC + (size_t)b * IPB * EDIM + d;
    float acc = 0.f;
    #pragma unroll 8
    for (int i = 0; i < IPB; ++i) acc += (float)base[(size_t)i * EDIM];
    u[b * EDIM + d] = acc;
}

// ---------------------------------------------------------------------------
// Kernel 3: one attention hop. One block (256 threads = 8 waves) per batch.
// Phase A: load u_b into LDS.  Phase B: p[i] = embA_i . u_b (wave per item).
// Phase C: shift = mean(relu(p)); w = softmax(p - shift) in LDS.
// Phase D: o = sum_i w_i * embC_i ; u = relu(u + o).
// ---------------------------------------------------------------------------
__global__ void hop_kernel(const _Float16* __restrict__ embA,
                           const _Float16* __restrict__ embC,
                           float* __restrict__ u) {
    __shared__ float u_s[EDIM];
    __shared__ float p_s[IPB];
    __shared__ float red_s[8];
    __shared__ float shift_s, part_s;

    const int b = blockIdx.x;
    const int tid = threadIdx.x;
    const int wave = tid >> 5;
    const int lane = tid & 31;

    u_s[tid]       = u[b * EDIM + tid];
    u_s[tid + 256] = u[b * EDIM + tid + 256];
    __syncthreads();

    // ---- Phase B: scores ----
    const size_t seg = (size_t)b * IPB;
    for (int it = wave; it < IPB; it += 8) {
        const _Float16* row = embA + (seg + it) * EDIM + lane * 16;
        if (it + 8 < IPB)
            __builtin_prefetch(embA + (seg + it + 8) * EDIM + lane * 16, 0, 1);
        h8 v0 = *reinterpret_cast<const h8*>(row);
        h8 v1 = *reinterpret_cast<const h8*>(row + 8);
        float acc = 0.f;
        #pragma unroll
        for (int j = 0; j < 8; ++j) acc += (float)v0[j] * u_s[lane * 16 + j];
        #pragma unroll
        for (int j = 0; j < 8; ++j) acc += (float)v1[j] * u_s[lane * 16 + 8 + j];
        acc = wave_sum(acc);
        if (lane == 0) p_s[it] = acc;
    }
    __syncthreads();

    // ---- Phase C: segment softmax with relu-mean shift ----
    const float pv = p_s[tid];
    float rv = wave_sum(fmaxf(pv, 0.f));
    if (lane == 0) red_s[wave] = rv;
    __syncthreads();
    if (tid == 0) {
        float s = 0.f;
        #pragma unroll
        for (int i = 0; i < 8; ++i) s += red_s[i];
        shift_s = s * (1.0f / (float)IPB);
    }
    __syncthreads();
    const float e = expf(pv - shift_s);
    float er = wave_sum(e);
    if (lane == 0) red_s[wave] = er;
    __syncthreads();
    if (tid == 0) {
        float s = 0.f;
        #pragma unroll
        for (int i = 0; i < 8; ++i) s += red_s[i];
        part_s = s;
    }
    __syncthreads();
    p_s[tid] = e / part_s;          // softmax weights, kept in LDS
    __syncthreads();

    // ---- Phase D: o = sum_i w_i * embC_i ; u = relu(u + o) ----
    const int d0 = tid * 2;         // thread owns dims d0, d0+1
    float o0 = 0.f, o1 = 0.f;
    const _Float16* cbase = embC + seg * EDIM + d0;
    #pragma unroll 8
    for (int i = 0; i < IPB; ++i) {
        const float wi = p_s[i];
        h2 hv = *reinterpret_cast<const h2*>(cbase + (size_t)i * EDIM);
        o0 += wi * (float)hv[0];
        o1 += wi * (float)hv[1];
    }
    u[b * EDIM + d0]     = fmaxf(u_s[d0] + o0, 0.f);
    u[b * EDIM + d0 + 1] = fmaxf(u_s[d0 + 1] + o1, 0.f);
}

// ---------------------------------------------------------------------------
// Kernel 4: head GEMM via V_WMMA_F32_16X16X4_F32 (exact f32 math).
// One wave per 16x16 tile of logits[256,64]; K = 512 in steps of 4.
// A layout (16x4 f32): lanes 0-15 -> K=0/1, lanes 16-31 -> K=2/3, M = lane&15.
// B layout (4x16 f32): same K split, N = lane&15.
// C/D: VGPR r = row (m0 + r + 8*(lane>=16)), col = n0 + (lane&15).
// ---------------------------------------------------------------------------
__global__ void head_wmma_kernel(const float* __restrict__ u,
                                 const float* __restrict__ W,
                                 float* __restrict__ logits) {
    const int m0 = blockIdx.x * 16;
    const int n0 = blockIdx.y * 16;
    const int lane = threadIdx.x;
    const int half = lane >> 4;
    const int mn = lane & 15;       // row for A, col for B
    const int kk = half * 2;        // K sub-offset for this lane group

    const float* arow = u + (size_t)(m0 + mn) * EDIM;
    v8f c = {};
    for (int k0 = 0; k0 < EDIM; k0 += 4) {
        v2f a, bm;
        a[0] = arow[k0 + kk];
        a[1] = arow[k0 + kk + 1];
        bm[0] = W[(size_t)(k0 + kk) * NACT + n0 + mn];
        bm[1] = W[(size_t)(k0 + kk + 1) * NACT + n0 + mn];
        c = __builtin_amdgcn_wmma_f32_16x16x4_f32(
                /*neg_a=*/false, a, /*neg_b=*/false, bm,
                /*c_mod=*/(short)0, c, /*reuse_a=*/false, /*reuse_b=*/false);
    }
    #pragma unroll
    for (int r = 0; r < 8; ++r)
        logits[(size_t)(m0 + r + half * 8) * NACT + n0 + mn] = c[r];
}

// ---------------------------------------------------------------------------
// Kernel 5: out = log_softmax(logits + bias) per row of 64. One wave per row.
// ---------------------------------------------------------------------------
__global__ void log_softmax_kernel(const float* __restrict__ logits,
                                   const float* __restrict__ bias,
                                   float* __restrict__ out) {
    const int row = blockIdx.x;
    const int t = threadIdx.x;      // 0..31, owns cols t and t+32
    float v0 = logits[row * NACT + t]      + bias[t];
    float v1 = logits[row * NACT + t + 32] + bias[t + 32];
    float m = fmaxf(v0, v1);
    #pragma unroll
    for (int off = 16; off > 0; off >>= 1) m = fmaxf(m, __shfl_xor(m, off, 32));
    float s = expf(v0 - m) + expf(v1 - m);
    #pragma unroll
    for (int off = 16; off > 0; off >>= 1) s += __shfl_xor(s, off, 32);
    const float lse = m + logf(s);
    out[row * NACT + t]      = v0 - lse;
    out[row * NACT + t + 32] = v1 - lse;
}

// ---------------------------------------------------------------------------
extern "C" void kernel_launch(void* const* d_in, const int* in_sizes, int n_in,
                              void* d_out, int out_size, void* d_ws, size_t ws_size,
                              hipStream_t stream) {
    // setup_inputs() order: x, item_starts, batch_idx, batch_len,
    //                       A_weight, C_weight, head_W, head_b
    const int*   x      = (const int*)  d_in[0];
    const float* Aw     = (const float*)d_in[4];
    const float* Cw     = (const float*)d_in[5];
    const float* headW  = (const float*)d_in[6];
    const float* headb  = (const float*)d_in[7];
    float*       out    = (float*)d_out;

    // Workspace layout (~135 MB): embA/embC in f16, u and logits in f32.
    _Float16* embA   = (_Float16*)d_ws;
    _Float16* embC   = embA + (size_t)NITEMS * EDIM;
    float*    u      = (float*)(embC + (size_t)NITEMS * EDIM);
    float*    logits = u + (size_t)BATCHES * EDIM;

    // 1) Embedding bags (one wave per item; 8 waves per block)
    embed_bag_kernel<<<NITEMS / 8, 256, 0, stream>>>(x, Aw, Cw, embA, embC);

    // 2) u = segment_sum(embC)
    u_init_kernel<<<BATCHES, EDIM, 0, stream>>>(embC, u);

    // 3) Three attention hops (one block per batch)
    for (int h = 0; h < NHOPS; ++h)
        hop_kernel<<<BATCHES, 256, 0, stream>>>(embA, embC, u);

    // 4) Head GEMM with f32 WMMA
    head_wmma_kernel<<<dim3(BATCHES / 16, NACT / 16), 32, 0, stream>>>(u, headW, logits);

    // 5) log_softmax + bias
    log_softmax_kernel<<<BATCHES, 32, 0, stream>>>(logits, headb, out);
}